// SpGraphTransAttentionLayer_5394478923812
// MI455X (gfx1250) — compile-verified
//
#include <hip/hip_runtime.h>
#include <hip/hip_bf16.h>

// ---------------------------------------------------------------------------
// SpGraphTransAttentionLayer for MI455X (gfx1250, wave32)
//   1) Q/K projections via V_WMMA_F32_16X16X4_F32 (fp32 WMMA -> fp32-exact)
//   2) per-(edge,head) 16-d dot products (L2-resident gathers; Q+K = 51 MB
//      fits the 192 MB L2, so gathers never hit HBM after the GEMM)
//   3) segment softmax via keyed atomicMax + f32 atomicAdd
// ---------------------------------------------------------------------------

typedef float v2f __attribute__((ext_vector_type(2)));
typedef float v8f __attribute__((ext_vector_type(8)));

#define IN_F   128
#define ATT_D  128
#define HEADS  8
#define D_K    16

// ---------------- Kernel 1: Q/K projection with fp32 WMMA -------------------
// Block = 256 threads = 8 waves; one block handles a 16-node tile.
// Wave w computes output-feature tile [16w, 16w+16) for both Wq and Wk.
// A operand (16x4 f32): lane<16 -> M=lane, VGPR0=K(2*half), VGPR1=K(2*half+1).
// B operand (4x16 f32): mirror layout with N as the lane dim -> a row pair of
// W (row-major [out, in]) loads directly as float2.
// C/D (16x16 f32): VGPR v -> M = v + 8*half, N = lane&15.
__global__ __launch_bounds__(256)
void qk_proj_wmma(const float* __restrict__ x,
                  const float* __restrict__ Wq, const float* __restrict__ bq,
                  const float* __restrict__ Wk, const float* __restrict__ bk,
                  float* __restrict__ Qout, float* __restrict__ Kout, int N)
{
    __shared__ float xs[16][132];               // pad 128->132: conflict-free
    const int tid      = threadIdx.x;
    const int nodeBase = blockIdx.x * 16;

    const int wave  = tid >> 5;
    const int lane  = tid & 31;
    const int half  = lane >> 4;     // 0: lanes 0-15, 1: lanes 16-31
    const int lmod  = lane & 15;
    const int oBase = wave * 16;     // output-feature tile base

    // Warm L2/L0 for this wave's W rows (global_prefetch_b8).
    __builtin_prefetch(Wq + (size_t)(oBase + lmod) * IN_F, 0, 3);
    __builtin_prefetch(Wk + (size_t)(oBase + lmod) * IN_F, 0, 3);

    // Stage the 16x128 x-tile into LDS (uniform loop -> no EXEC divergence).
    for (int i = tid; i < 16 * IN_F; i += 256) {
        int r = i >> 7, c = i & 127;
        int row = nodeBase + r;
        xs[r][c] = (row < N) ? x[(size_t)row * IN_F + c] : 0.0f;
    }
    __syncthreads();

    const float2* xrow = (const float2*)(&xs[lmod][0]);   // 8B-aligned rows
    const bool fullTile = (nodeBase + 16 <= N);           // uniform per block

    for (int m = 0; m < 2; ++m) {
        const float* W    = (m == 0) ? Wq : Wk;
        const float* bvec = (m == 0) ? bq : bk;
        float*       Out  = (m == 0) ? Qout : Kout;
        const float2* wrow = (const float2*)(W + (size_t)(oBase + lmod) * IN_F);

        v8f acc = {0.f, 0.f, 0.f, 0.f, 0.f, 0.f, 0.f, 0.f};
        #pragma unroll 8
        for (int kk = 0; kk < IN_F / 4; ++kk) {      // K advances by 4 / wmma
            float2 av = xrow[2 * kk + half];          // x[m][4kk+2h .. +1]
            float2 wv = wrow[2 * kk + half];          // W[o][4kk+2h .. +1]
            v2f a; a[0] = av.x; a[1] = av.y;
            v2f b; b[0] = wv.x; b[1] = wv.y;
            acc = __builtin_amdgcn_wmma_f32_16x16x4_f32(
                      false, a, false, b, (short)0, acc, false, false);
        }

        const float bias = bvec[oBase + lmod];
        if (fullTile) {
            // Straight-line stores; compiler can clause these.
            float* p = Out + (size_t)(nodeBase + 8 * half) * ATT_D + oBase + lmod;
            #pragma unroll
            for (int v = 0; v < 8; ++v)
                p[(size_t)v * ATT_D] = acc[v] + bias;
        } else {
            #pragma unroll
            for (int v = 0; v < 8; ++v) {
                int row = nodeBase + v + 8 * half;
                if (row < N)
                    Out[(size_t)row * ATT_D + oBase + lmod] = acc[v] + bias;
            }
        }
    }
}

// ---------------- Segment-softmax helpers -----------------------------------
// Monotonic float <-> uint key so unsigned atomicMax == float max.
__device__ __forceinline__ unsigned fkey(float x) {
    unsigned u = __float_as_uint(x);
    return (u & 0x80000000u) ? ~u : (u | 0x80000000u);
}
__device__ __forceinline__ float fdec(unsigned k) {
    return __uint_as_float((k & 0x80000000u) ? (k & 0x7FFFFFFFu) : ~k);
}

__global__ void init_seg(unsigned* __restrict__ segmax,
                         float* __restrict__ segsum, int n)
{
    int i = blockIdx.x * blockDim.x + threadIdx.x;
    if (i < n) { segmax[i] = 0u; segsum[i] = 0.0f; }  // key 0 < any finite key
}

// ---------------- Kernel 2: scores + segment max ----------------------------
__global__ __launch_bounds__(256)
void edge_scores(const float* __restrict__ Q, const float* __restrict__ K,
                 const int* __restrict__ esrc, const int* __restrict__ edst,
                 float* __restrict__ score, unsigned* __restrict__ segmax,
                 long total)
{
    long idx = (long)blockIdx.x * blockDim.x + threadIdx.x;
    if (idx >= total) return;
    int e = (int)(idx >> 3), h = (int)(idx & 7);
    int s = esrc[e], d = edst[e];
    const float4* qp = (const float4*)(Q + (size_t)s * ATT_D + h * D_K);
    const float4* kp = (const float4*)(K + (size_t)d * ATT_D + h * D_K);
    float acc = 0.f;
    #pragma unroll
    for (int i = 0; i < 4; ++i) {
        float4 a = qp[i], b = kp[i];
        acc += a.x * b.x + a.y * b.y + a.z * b.z + a.w * b.w;
    }
    acc *= 0.25f;                                  // 1/sqrt(D_K)
    score[idx] = acc;
    atomicMax(&segmax[(size_t)d * HEADS + h], fkey(acc));
}

// ---------------- Kernel 3: exp + segment sum -------------------------------
__global__ __launch_bounds__(256)
void edge_exp(const int* __restrict__ edst, float* __restrict__ score,
              const unsigned* __restrict__ segmax, float* __restrict__ segsum,
              long total)
{
    long idx = (long)blockIdx.x * blockDim.x + threadIdx.x;
    if (idx >= total) return;
    int e = (int)(idx >> 3), h = (int)(idx & 7);
    int d = edst[e];
    float m  = fdec(segmax[(size_t)d * HEADS + h]);
    float ex = expf(score[idx] - m);
    score[idx] = ex;
    atomicAdd(&segsum[(size_t)d * HEADS + h], ex);   // global_atomic_add_f32
}

// ---------------- Kernel 4: normalize ---------------------------------------
__global__ __launch_bounds__(256)
void edge_norm(const int* __restrict__ edst, float* __restrict__ score,
               const float* __restrict__ segsum, long total)
{
    long idx = (long)blockIdx.x * blockDim.x + threadIdx.x;
    if (idx >= total) return;
    int e = (int)(idx >> 3), h = (int)(idx & 7);
    int d = edst[e];
    float s = segsum[(size_t)d * HEADS + h];
    score[idx] = score[idx] / (s + 1e-16f);
}

// ---------------- Launch ----------------------------------------------------
extern "C" void kernel_launch(void* const* d_in, const int* in_sizes, int n_in,
                              void* d_out, int out_size, void* d_ws, size_t ws_size,
                              hipStream_t stream)
{
    const float* x    = (const float*)d_in[0];
    const int*   edge = (const int*)  d_in[1];   // [2, E] int32 per harness
    const float* Wq   = (const float*)d_in[2];
    const float* bq   = (const float*)d_in[3];
    const float* Wk   = (const float*)d_in[4];
    const float* bk   = (const float*)d_in[5];

    const int N = in_sizes[0] / IN_F;
    const int E = in_sizes[1] / 2;
    const int* esrc = edge;
    const int* edst = edge + E;
    float* out = (float*)d_out;                  // [E, HEADS] in place

    // Workspace layout: Q | K | segmax(u32) | segsum(f32)
    float*    Q      = (float*)d_ws;
    float*    K      = Q + (size_t)N * ATT_D;
    unsigned* segmax = (unsigned*)(K + (size_t)N * ATT_D);
    float*    segsum = (float*)(segmax + (size_t)N * HEADS);

    const int nSeg = N * HEADS;
    init_seg<<<(nSeg + 255) / 256, 256, 0, stream>>>(segmax, segsum, nSeg);

    const int nTiles = (N + 15) / 16;
    qk_proj_wmma<<<nTiles, 256, 0, stream>>>(x, Wq, bq, Wk, bk, Q, K, N);

    const long total  = (long)E * HEADS;
    const int  blocks = (int)((total + 255) / 256);
    edge_scores<<<blocks, 256, 0, stream>>>(Q, K, esrc, edst, out, segmax, total);
    edge_exp   <<<blocks, 256, 0, stream>>>(edst, out, segmax, segsum, total);
    edge_norm  <<<blocks, 256, 0, stream>>>(edst, out, segsum, total);
}